// AttentionLayer_81157702025328
// MI455X (gfx1250) — compile-verified
//
#include <hip/hip_runtime.h>
#include <hip/hip_bf16.h>
#include <math.h>

typedef __attribute__((ext_vector_type(2))) float v2f;
typedef __attribute__((ext_vector_type(8))) float v8f;

static constexpr int kH = 1024;   // hidden dim
static constexpr int kB = 64;     // batch

// ---------------------------------------------------------------------------
// Kernel 1: v[b][h] = sum_k hidden[b][k] * W[k][h]   (v = hidden @ W)
// One wave per 16x16 tile of v, V_WMMA_F32_16X16X4_F32 over K in steps of 4.
// A (16x4) layout: lanes 0-15 hold M=0..15 / K=0,1 ; lanes 16-31 hold K=2,3.
// B (4x16) layout: rows striped across lanes; VGPR0 = K0 (lo half)/K2 (hi),
//                  VGPR1 = K1/K3.
// C/D (16x16): 8 VGPRs, lane<16 -> M=r, lane>=16 -> M=r+8, N=lane&15.
// ---------------------------------------------------------------------------
__global__ __launch_bounds__(32) void gemm_vb(const float* __restrict__ hidden,
                                              const float* __restrict__ W,
                                              float* __restrict__ v) {
  const int lane  = threadIdx.x;              // 0..31, EXEC all ones
  const int tile  = blockIdx.x;               // 0..255
  const int tm    = (tile & 3) * 16;          // b-tile base (64/16 = 4 tiles)
  const int tn    = (tile >> 2) * 16;         // h-tile base (1024/16 = 64)
  const int mr    = lane & 15;                // M (A) or N (B) within tile
  const int khalf = (lane >> 4) * 2;          // 0 for lanes<16, 2 otherwise

  v8f acc = {};
  const float* Arow = hidden + (tm + mr) * kH;    // hidden[b][*]
  const float* Bcol = W + tn + mr;                // W[*][h]

  for (int k0 = 0; k0 < kH; k0 += 4) {
    v2f a, b;
    a.x = Arow[k0 + khalf];
    a.y = Arow[k0 + khalf + 1];
    b.x = Bcol[(size_t)(k0 + khalf) * kH];
    b.y = Bcol[(size_t)(k0 + khalf + 1) * kH];
    acc = __builtin_amdgcn_wmma_f32_16x16x4_f32(
        /*neg_a=*/false, a, /*neg_b=*/false, b,
        /*c_mod=*/(short)0, acc, /*reuse_a=*/false, /*reuse_b=*/false);
  }

  const int n     = tn + mr;
  const int mbase = tm + (lane >> 4) * 8;
#pragma unroll
  for (int r = 0; r < 8; ++r) {
    v[(size_t)(mbase + r) * kH + n] = acc[r];
  }
}

// ---------------------------------------------------------------------------
// Kernel 2: c[b] = sum_k hidden[b][k] * bias[k]
// ---------------------------------------------------------------------------
__global__ __launch_bounds__(256) void bias_dot(const float* __restrict__ hidden,
                                                const float* __restrict__ bias,
                                                float* __restrict__ c) {
  __shared__ float red[256];
  const int b = blockIdx.x;
  const int t = threadIdx.x;
  float s = 0.f;
  for (int i = t; i < kH; i += 256) s += hidden[b * kH + i] * bias[i];
  red[t] = s;
  __syncthreads();
  for (int off = 128; off > 0; off >>= 1) {
    if (t < off) red[t] += red[t + off];
    __syncthreads();
  }
  if (t == 0) c[b] = red[0];
}

// ---------------------------------------------------------------------------
// Kernel 3 (bandwidth-bound main pass):
//   e[b*L + l] = v[b] . enc[l][b] + c[b]
// One wave per (l,b) pair; float4 loads -> 512B contiguous per wave per instr.
// ---------------------------------------------------------------------------
__global__ __launch_bounds__(256) void energies(const float* __restrict__ enc,
                                                const float* __restrict__ v,
                                                const float* __restrict__ c,
                                                float* __restrict__ e, int L) {
  const int wave = threadIdx.x >> 5;
  const int lane = threadIdx.x & 31;
  const int p = blockIdx.x * 8 + wave;   // flat (l,b) index
  const int l = p >> 6;
  const int b = p & 63;

  const float4* ep = (const float4*)(enc + ((size_t)l * kB + b) * kH);
  const float4* vp = (const float4*)(v + (size_t)b * kH);

  float acc = 0.f;
#pragma unroll
  for (int i = 0; i < kH / 128; ++i) {   // 8 iterations of float4 per lane
    float4 x = ep[lane + i * 32];
    float4 y = vp[lane + i * 32];
    acc += x.x * y.x + x.y * y.y + x.z * y.z + x.w * y.w;
  }
  for (int off = 16; off > 0; off >>= 1)
    acc += __shfl_down(acc, off, 32);
  if (lane == 0) e[(size_t)b * L + l] = acc + c[b];
}

// ---------------------------------------------------------------------------
// Kernel 4: softmax over the BATCH axis (dim=0 of e[B,L]), out[b][0][l].
// Thread per l; all 64 batch reads/writes at fixed b are coalesced across l.
// ---------------------------------------------------------------------------
__global__ __launch_bounds__(256) void softmax_b(const float* __restrict__ e,
                                                 float* __restrict__ out, int L) {
  const int l = blockIdx.x * 256 + threadIdx.x;
  if (l >= L) return;
  float vals[kB];
  float m = -INFINITY;
#pragma unroll
  for (int b = 0; b < kB; ++b) {
    vals[b] = e[(size_t)b * L + l];
    m = fmaxf(m, vals[b]);
  }
  float s = 0.f;
#pragma unroll
  for (int b = 0; b < kB; ++b) {
    vals[b] = __expf(vals[b] - m);
    s += vals[b];
  }
  const float inv = 1.0f / s;
#pragma unroll
  for (int b = 0; b < kB; ++b)
    out[(size_t)b * L + l] = vals[b] * inv;
}

// ---------------------------------------------------------------------------
extern "C" void kernel_launch(void* const* d_in, const int* in_sizes, int n_in,
                              void* d_out, int out_size, void* d_ws, size_t ws_size,
                              hipStream_t stream) {
  const float* hidden = (const float*)d_in[0];   // [1, 64, 1024]
  const float* enc    = (const float*)d_in[1];   // [L, 64, 1024]
  const float* W      = (const float*)d_in[2];   // [1024, 1024] (out,in)
  const float* bias   = (const float*)d_in[3];   // [1024]
  float* out = (float*)d_out;                    // [64, 1, L]

  const int L = in_sizes[1] / (kB * kH);         // 2048 for reference shapes

  float* e = (float*)d_ws;                       // B*L floats
  float* v = e + (size_t)kB * L;                 // B*H floats
  float* c = v + (size_t)kB * kH;                // B floats

  gemm_vb<<<(kB / 16) * (kH / 16), 32, 0, stream>>>(hidden, W, v);
  bias_dot<<<kB, 256, 0, stream>>>(hidden, bias, c);
  energies<<<(L * kB) / 8, 256, 0, stream>>>(enc, v, c, e, L);
  softmax_b<<<(L + 255) / 256, 256, 0, stream>>>(e, out, L);
}